// Block_40716289966389
// MI455X (gfx1250) — compile-verified
//
#include <hip/hip_runtime.h>

typedef __attribute__((ext_vector_type(2))) float v2f;
typedef __attribute__((ext_vector_type(8))) float v8f;

#define V_N     196608
#define K_CH    4
#define F_IO    64
#define KK      256        // K_CH * F_IO : GEMM reduction dim
#define NGROUPS 8
#define GN_EPS  1e-5f

// ---------------------------------------------------------------------------
// 1) zero the GroupNorm stats accumulators (16 floats in workspace)
// ---------------------------------------------------------------------------
__global__ void zero_stats_kernel(float* __restrict__ stats) {
    if (threadIdx.x < 2 * NGROUPS) stats[threadIdx.x] = 0.0f;
}

// ---------------------------------------------------------------------------
// 2) init a Chebyshev slice of cheb[V][256]:
//    dst slice = alpha * src slice   (alpha==0 -> pure zero init, src ignored)
//    All addressing in float4 units. cheb row = 64 float4.
// ---------------------------------------------------------------------------
__global__ __launch_bounds__(256)
void init_slice_kernel(float* __restrict__ cheb, const float* __restrict__ src,
                       int dst_off_f4, int src_stride_f4, int src_off_f4,
                       float alpha, int total_f4) {
    int t = blockIdx.x * blockDim.x + threadIdx.x;
    if (t >= total_f4) return;
    int v  = t >> 4;          // vertex
    int f4 = t & 15;          // float4 within 64-ch slice
    float4 o = make_float4(0.f, 0.f, 0.f, 0.f);
    if (alpha != 0.0f) {
        const float4* s = (const float4*)src;
        float4 x = s[(long long)v * src_stride_f4 + src_off_f4 + f4];
        o = make_float4(alpha * x.x, alpha * x.y, alpha * x.z, alpha * x.w);
    }
    ((float4*)cheb)[(long long)v * 64 + dst_off_f4 + f4] = o;
}

// ---------------------------------------------------------------------------
// 3) SpMM scatter: cheb[:,dst] += scale * vals[e] * cheb[cols[e], src]
//    thread t -> (edge, feature). A wave32 spans 32 consecutive t, so the
//    edge index e = t>>6 is wave-uniform: readfirstlane scalarizes the
//    rows/cols/vals descriptor loads into SMEM (s_load) traffic, and the
//    per-lane gather of x[col,f] / atomic scatter to y[row,f] coalesce.
// ---------------------------------------------------------------------------
__global__ __launch_bounds__(256)
void spmm_edge_kernel(float* __restrict__ cheb,
                      const int* __restrict__ rows, const int* __restrict__ cols,
                      const float* __restrict__ vals,
                      int src_off, int dst_off, float scale, int E) {
    int t = blockIdx.x * blockDim.x + threadIdx.x;
    int e = __builtin_amdgcn_readfirstlane(t >> 6);   // wave-uniform edge
    int f = t & 63;
    if (e >= E) return;
    int   r = rows[e];                                 // scalar loads
    int   c = cols[e];
    float w = vals[e] * scale;
    float xv = cheb[(long long)c * KK + src_off + f];
    atomicAdd(&cheb[(long long)r * KK + dst_off + f], w * xv);
}

// ---------------------------------------------------------------------------
// 4) Dense contraction with WMMA:
//    y[V,64] = cheb[V,256] @ W[256,64] + bias, fused GroupNorm partial stats.
//    256 threads = 8 wave32; each wave: 16 rows x 64 cols via 4 f32 acc tiles.
//    W (64 KB) is staged into LDS with GLOBAL_LOAD_ASYNC_TO_LDS_B128
//    (ASYNCcnt path, no VGPR round-trip), each wave waits its own ASYNCcnt
//    before the workgroup barrier.
// ---------------------------------------------------------------------------
__global__ __launch_bounds__(256)
void cheb_gemm_kernel(const float* __restrict__ cheb,
                      const float* __restrict__ Wt,     // [256][64] row-major
                      const float* __restrict__ bias,   // [64]
                      float* __restrict__ y,            // [V][64]
                      float* __restrict__ stats) {      // [8][2] sum,sumsq
    __shared__ float Wlds[KK * F_IO];                   // 64 KB in 320 KB WGP LDS

    const int tid = threadIdx.x;

    // --- async stage: 4096 b128 transfers, 16 per thread --------------------
    {
        const unsigned lds_base = (unsigned)(unsigned long long)(&Wlds[0]);
        #pragma unroll
        for (int i = 0; i < 16; ++i) {
            const int idx = tid + i * 256;                       // float4 index
            const unsigned long long gaddr =
                (unsigned long long)(const void*)(((const float4*)Wt) + idx);
            const unsigned lds_off = lds_base + (unsigned)idx * 16u;
            asm volatile("global_load_async_to_lds_b128 %0, %1, off"
                         :: "v"(lds_off), "v"(gaddr)
                         : "memory");
        }
        asm volatile("s_wait_asynccnt 0x0" ::: "memory");
    }
    __syncthreads();

    const int wave = tid >> 5;
    const int lane = tid & 31;
    const int half = lane >> 4;
    const int ln   = lane & 15;
    const long long row0 = ((long long)blockIdx.x * 8 + wave) * 16;

    v8f acc0 = {}, acc1 = {}, acc2 = {}, acc3 = {};
    const float* arow = cheb + (row0 + ln) * KK;
    __builtin_prefetch(arow, 0, 0);                    // global_prefetch_b8
    __builtin_prefetch(arow + 128, 0, 0);

    for (int k0 = 0; k0 < KK; k0 += 4) {
        const int ka = k0 + 2 * half;
        v2f a;
        a.x = arow[ka];
        a.y = arow[ka + 1];
        const float* wrow0 = &Wlds[ka * F_IO + ln];
        const float* wrow1 = &Wlds[(ka + 1) * F_IO + ln];
        v2f b;
        b.x = wrow0[0];  b.y = wrow1[0];
        acc0 = __builtin_amdgcn_wmma_f32_16x16x4_f32(false, a, false, b, (short)0, acc0, false, false);
        b.x = wrow0[16]; b.y = wrow1[16];
        acc1 = __builtin_amdgcn_wmma_f32_16x16x4_f32(false, a, false, b, (short)0, acc1, false, false);
        b.x = wrow0[32]; b.y = wrow1[32];
        acc2 = __builtin_amdgcn_wmma_f32_16x16x4_f32(false, a, false, b, (short)0, acc2, false, false);
        b.x = wrow0[48]; b.y = wrow1[48];
        acc3 = __builtin_amdgcn_wmma_f32_16x16x4_f32(false, a, false, b, (short)0, acc3, false, false);
    }

    // epilogue: add bias, store y, wave-reduce GroupNorm stats
    v8f accs[4] = {acc0, acc1, acc2, acc3};
    #pragma unroll
    for (int j = 0; j < 4; ++j) {
        const int n  = j * 16 + ln;        // output channel
        const float bz = bias[n];
        float s = 0.f, ss = 0.f;
        #pragma unroll
        for (int r = 0; r < 8; ++r) {
            float d = accs[j][r] + bz;
            y[(row0 + r + 8 * half) * F_IO + n] = d;
            s += d; ss += d * d;
        }
        // reduce across the 8 lanes of this channel-group and the two halves
        s  += __shfl_xor(s, 1, 32);  ss += __shfl_xor(ss, 1, 32);
        s  += __shfl_xor(s, 2, 32);  ss += __shfl_xor(ss, 2, 32);
        s  += __shfl_xor(s, 4, 32);  ss += __shfl_xor(ss, 4, 32);
        s  += __shfl_xor(s, 16, 32); ss += __shfl_xor(ss, 16, 32);
        if (half == 0 && (ln == 0 || ln == 8)) {
            int g = n >> 3;
            atomicAdd(&stats[2 * g + 0], s);
            atomicAdd(&stats[2 * g + 1], ss);
        }
    }
}

// ---------------------------------------------------------------------------
// 5) finalize: GroupNorm affine + PReLU, in place on y (= d_out)
// ---------------------------------------------------------------------------
__global__ __launch_bounds__(256)
void groupnorm_prelu_kernel(float* __restrict__ y, const float* __restrict__ stats,
                            const float* __restrict__ gamma, const float* __restrict__ beta,
                            const float* __restrict__ prelu_a, int total) {
    __shared__ float s_mean[NGROUPS], s_inv[NGROUPS];
    if (threadIdx.x < NGROUPS) {
        const float cnt = (float)V_N * 8.0f;
        float m  = stats[2 * threadIdx.x + 0] / cnt;
        float vr = stats[2 * threadIdx.x + 1] / cnt - m * m;
        s_mean[threadIdx.x] = m;
        s_inv[threadIdx.x]  = rsqrtf(vr + GN_EPS);
    }
    __syncthreads();
    const float a = prelu_a[0];
    for (int t = blockIdx.x * blockDim.x + threadIdx.x; t < total;
         t += gridDim.x * blockDim.x) {
        int c = t & 63;
        int g = c >> 3;
        float v  = y[t];
        float yn = (v - s_mean[g]) * s_inv[g] * gamma[c] + beta[c];
        y[t] = (yn >= 0.f) ? yn : a * yn;
    }
}

// ---------------------------------------------------------------------------
extern "C" void kernel_launch(void* const* d_in, const int* in_sizes, int n_in,
                              void* d_out, int out_size, void* d_ws, size_t ws_size,
                              hipStream_t stream) {
    const float* x        = (const float*)d_in[0];   // [1,V,64]
    const int*   lap_rows = (const int*)d_in[1];     // [E]
    const int*   lap_cols = (const int*)d_in[2];     // [E]
    const float* lap_vals = (const float*)d_in[3];   // [E]
    const float* weight   = (const float*)d_in[4];   // [K,64,64] -> [256,64]
    const float* bias     = (const float*)d_in[5];   // [64]
    const float* gamma    = (const float*)d_in[6];   // [64]
    const float* beta     = (const float*)d_in[7];   // [64]
    const float* prelu_a  = (const float*)d_in[8];   // [1]
    const int E = in_sizes[1];

    float* cheb  = (float*)d_ws;                                 // [V][256]
    float* stats = (float*)((char*)d_ws + (size_t)V_N * KK * 4); // [16]
    float* y     = (float*)d_out;                                // [V][64]

    const int total_f4   = V_N * 16;                 // float4s per 64-ch slice
    const int init_grid  = (total_f4 + 255) / 256;
    const int spmm_grid  = (E * 64 + 255) / 256;

    zero_stats_kernel<<<1, 32, 0, stream>>>(stats);

    // x0 = x  (slice 0)
    init_slice_kernel<<<init_grid, 256, 0, stream>>>(cheb, x, 0, 16, 0, 1.0f, total_f4);
    // x1 = L x0 : slice1 = 0, then scatter vals * x0
    init_slice_kernel<<<init_grid, 256, 0, stream>>>(cheb, x, 16, 16, 0, 0.0f, total_f4);
    spmm_edge_kernel<<<spmm_grid, 256, 0, stream>>>(cheb, lap_rows, lap_cols, lap_vals,
                                                    0, 64, 1.0f, E);
    // x2 = 2 L x1 - x0 : slice2 = -slice0, then scatter 2*vals * x1
    init_slice_kernel<<<init_grid, 256, 0, stream>>>(cheb, cheb, 32, 64, 0, -1.0f, total_f4);
    spmm_edge_kernel<<<spmm_grid, 256, 0, stream>>>(cheb, lap_rows, lap_cols, lap_vals,
                                                    64, 128, 2.0f, E);
    // x3 = 2 L x2 - x1 : slice3 = -slice1, then scatter 2*vals * x2
    init_slice_kernel<<<init_grid, 256, 0, stream>>>(cheb, cheb, 48, 64, 16, -1.0f, total_f4);
    spmm_edge_kernel<<<spmm_grid, 256, 0, stream>>>(cheb, lap_rows, lap_cols, lap_vals,
                                                    128, 192, 2.0f, E);

    // y = cheb @ W + bias  (WMMA f32), fused GroupNorm stats
    cheb_gemm_kernel<<<V_N / 128, 256, 0, stream>>>(cheb, weight, bias, y, stats);

    // GroupNorm affine + PReLU in place
    groupnorm_prelu_kernel<<<1024, 256, 0, stream>>>(y, stats, gamma, beta, prelu_a,
                                                     V_N * F_IO);
}